// CentroidInstanceLoss_24060406792992
// MI455X (gfx1250) — compile-verified
//
#include <hip/hip_runtime.h>
#include <cstdint>

// ---------------------------------------------------------------------------
// CentroidInstanceLoss for MI455X (gfx1250)
//   N=1e6 points, D=32, B=8 subbatches, L=64 labels, S=512 segments.
//   Memory-bound: 2 streaming passes over 128 MB => ~11 us floor @ 23.3 TB/s.
//   CDNA5 features used: 320KB-LDS segment accumulators (ds_add_f32),
//   TDM tensor_load_to_lds (+LDS padding for bank-conflict-free reads),
//   s_wait_tensorcnt, global_prefetch_b8.
// ---------------------------------------------------------------------------

#define NPTS   1000000
#define DIM    32
#define NB     8
#define NL     64
#define NSEG   512              // NB*NL
#define DELTA_V 0.5f
#define DELTA_D 1.5f

// workspace layout (float indices)
#define WS_SUMS   0             // 16384 floats: segment sums -> centroids (in place)
#define WS_SCALE  16384         // 512  floats: pull scale = present ? 1/(M_b*N_m) : 0
#define WS_COUNTS 16896         // 512  floats
#define WS_PUSHB  17408         // 8    floats: per-subbatch push sums
#define WS_M      17416         // 8    floats: M per subbatch
#define WS_BCOUNT 17424         // 1
#define WS_PULL   17425         // 1
#define WS_TOTAL  17426

// LDS table in pull pass: 16896 source dwords + 1 pad dword per 32 => 17424
#define PULL_LDSW 17424

typedef unsigned int u32x4 __attribute__((ext_vector_type(4)));
typedef int          i32x4 __attribute__((ext_vector_type(4)));
typedef int          i32x8 __attribute__((ext_vector_type(8)));

// --- compile-time feedback: which TDM path is this toolchain taking? -------
#if __has_builtin(__builtin_amdgcn_tensor_load_to_lds)
#if defined(__clang_major__) && (__clang_major__ >= 23)
#pragma message("CIL-DIAG: TDM builtin PRESENT, using 6-arg form (clang>=23)")
#else
#pragma message("CIL-DIAG: TDM builtin PRESENT, using 5-arg form (clang<23)")
#endif
#else
#pragma message("CIL-DIAG: TDM builtin MISSING, cooperative-copy fallback")
#endif
#if __has_builtin(__builtin_amdgcn_s_wait_tensorcnt)
#pragma message("CIL-DIAG: s_wait_tensorcnt builtin PRESENT")
#else
#pragma message("CIL-DIAG: s_wait_tensorcnt builtin MISSING, inline asm")
#endif

// ---------------------------------------------------------------- init ------
__global__ __launch_bounds__(256) void cil_init(float* ws) {
    int i = blockIdx.x * 256 + threadIdx.x;
    if (i < WS_TOTAL) ws[i] = 0.0f;
}

// ------------------------------------------------------------- helpers ------
__device__ __forceinline__ void load_point(const float4* __restrict__ x4,
                                           size_t i, float c[DIM]) {
    const float4* p = x4 + i * 8;
#pragma unroll
    for (int k = 0; k < 8; ++k) {
        float4 t = p[k];
        c[4 * k + 0] = t.x; c[4 * k + 1] = t.y;
        c[4 * k + 2] = t.z; c[4 * k + 3] = t.w;
    }
}

__device__ __forceinline__ float inv_norm(const float c[DIM]) {
    float ss = 0.f;
#pragma unroll
    for (int d = 0; d < DIM; ++d) ss += c[d] * c[d];
    return 1.0f / (sqrtf(ss) + 1e-8f);
}

// ------------------------------------------------- pass 1: segment sums -----
// LDS accumulators: sums layout [d*512+seg] -> LDS bank = seg%64 (conflict-free
// for random segs). 66KB LDS/workgroup (needs CDNA5's 320KB WGP LDS).
__global__ __launch_bounds__(256) void cil_pass1(const float4* __restrict__ x4,
                                                 const int* __restrict__ lab,
                                                 const int* __restrict__ sb,
                                                 float* __restrict__ ws, int n) {
    __shared__ float lsum[NSEG * DIM];   // 64 KB
    __shared__ float lcnt[NSEG];         //  2 KB
    for (int j = threadIdx.x; j < NSEG * DIM; j += 256) lsum[j] = 0.f;
    for (int j = threadIdx.x; j < NSEG; j += 256) lcnt[j] = 0.f;
    __syncthreads();

    const int stride = gridDim.x * 256;
    for (int i = blockIdx.x * 256 + threadIdx.x; i < n; i += stride) {
        __builtin_prefetch((const void*)(x4 + (size_t)(i + stride) * 8), 0, 1);
        float c[DIM];
        load_point(x4, (size_t)i, c);
        float inv = inv_norm(c);
        int seg = sb[i] * NL + lab[i];
#pragma unroll
        for (int d = 0; d < DIM; ++d)
            atomicAdd(&lsum[d * NSEG + seg], c[d] * inv);   // ds_add_f32
        atomicAdd(&lcnt[seg], 1.0f);
    }
    __syncthreads();

    // flush block partials (L2-resident 66KB target, fp32 global atomics)
    float* gsums = ws + WS_SUMS;
    float* gcnt  = ws + WS_COUNTS;
    for (int j = threadIdx.x; j < NSEG * DIM; j += 256) {
        float v = lsum[j];
        if (v != 0.f) {
            int d = j >> 9, seg = j & (NSEG - 1);
            atomicAdd(&gsums[seg * DIM + d], v);   // global layout [seg][d]
        }
    }
    for (int j = threadIdx.x; j < NSEG; j += 256) {
        float v = lcnt[j];
        if (v != 0.f) atomicAdd(&gcnt[j], v);
    }
}

// ------------------------------------ centroids, M, Bcount, pull scales -----
__global__ __launch_bounds__(512) void cil_centroids(float* __restrict__ ws) {
    __shared__ int cntb[NB];
    int s = threadIdx.x;                        // 0..511 = one segment each
    float c = ws[WS_COUNTS + s];
    bool present = c > 0.f;
    float invc = present ? 1.0f / c : 0.f;
#pragma unroll
    for (int d = 0; d < DIM; ++d)               // sums -> mus (in place)
        ws[WS_SUMS + s * DIM + d] *= invc;
    if (s < NB) cntb[s] = 0;
    __syncthreads();
    if (present) atomicAdd(&cntb[s >> 6], 1);
    __syncthreads();
    float Mb = (float)cntb[s >> 6];
    ws[WS_SCALE + s] = present ? 1.0f / (Mb * c) : 0.0f;
    if (s < NB) ws[WS_M + s] = (float)cntb[s];
    if (s == 0) {
        int bc = 0;
        for (int b = 0; b < NB; ++b) bc += (cntb[b] > 0);
        ws[WS_BCOUNT] = (float)bc;
    }
}

// ------------------------------------------------ push: centroid pairs ------
// 32768 ordered pairs; 16 blocks per subbatch (grid 128 x 256 threads).
__global__ __launch_bounds__(256) void cil_push(float* __restrict__ ws) {
    __shared__ float red[256];
    int id = blockIdx.x * 256 + threadIdx.x;    // pair id in [0, 32768)
    int b  = id >> 12, l1 = (id >> 6) & 63, l2 = id & 63;
    float v = 0.f;
    if (l1 != l2) {
        bool p1 = ws[WS_COUNTS + b * NL + l1] > 0.f;
        bool p2 = ws[WS_COUNTS + b * NL + l2] > 0.f;
        if (p1 && p2) {
            const float* m1 = ws + WS_SUMS + (b * NL + l1) * DIM;
            const float* m2 = ws + WS_SUMS + (b * NL + l2) * DIM;
            float pd = 0.f;
#pragma unroll
            for (int d = 0; d < DIM; ++d) pd += fabsf(m1[d] - m2[d]);
            float h = fmaxf(2.0f * DELTA_D - pd, 0.f);
            v = h * h;
        }
    }
    red[threadIdx.x] = v; __syncthreads();
    for (int off = 128; off > 0; off >>= 1) {
        if (threadIdx.x < off) red[threadIdx.x] += red[threadIdx.x + off];
        __syncthreads();
    }
    if (threadIdx.x == 0) atomicAdd(&ws[WS_PUSHB + b], red[0]); // block is single-b
}

// --------------------------------------------------- pass 2: pull term ------
// Stage mus(16384)+scale(512) global->LDS with TDM. D# pad feature inserts one
// pad dword per 32 => LDS row stride 33 => mus reads hit 64 distinct banks.
__global__ __launch_bounds__(256) void cil_pull(const float4* __restrict__ x4,
                                                const int* __restrict__ lab,
                                                const int* __restrict__ sb,
                                                const float* __restrict__ ws,
                                                float* __restrict__ pull_acc,
                                                int n) {
    __shared__ float lds[PULL_LDSW];            // ~69.7 KB
    __shared__ float red[256];

#if __has_builtin(__builtin_amdgcn_tensor_load_to_lds)
    if (threadIdx.x < 32) {                     // first wave only (EXEC ignored by TDM)
        uint32_t lbase = (uint32_t)(uintptr_t)(&lds[0]);
        uint64_t ga = (uint64_t)(uintptr_t)(ws + WS_SUMS);
        // D# group0: count=1 | lds_addr | global_addr[56:0] | type=2
        u32x4 g0 = { 1u, lbase, (uint32_t)(ga & 0xFFFFFFFFu),
                     (uint32_t)((ga >> 32) & 0x01FFFFFFu) | 0x80000000u };
        // D# group1: data_size=4B, pad_enable, pad_interval=32dw, pad_amount=1dw
        //   tensor_dim0=32, tensor_dim1=528, tile=32x528, dim0_stride=32,
        //   dim1_stride=16896 (unused for 2D tile)
        i32x8 g1 = { (int)0x01120000u,
                     (int)(32u  << 16),
                     (int)(528u << 16),
                     (int)(32u  << 16),
                     (int)528,
                     (int)32,
                     (int)0x42000000u,
                     0 };
        i32x4 g2 = { 0, 0, 0, 0 };
        i32x4 g3 = { 0, 0, 0, 0 };
#if defined(__clang_major__) && (__clang_major__ >= 23)
        i32x8 g4 = { 0, 0, 0, 0, 0, 0, 0, 0 };
        __builtin_amdgcn_tensor_load_to_lds(g0, g1, g2, g3, g4, 0);
#else
        __builtin_amdgcn_tensor_load_to_lds(g0, g1, g2, g3, 0);
#endif
#if __has_builtin(__builtin_amdgcn_s_wait_tensorcnt)
        __builtin_amdgcn_s_wait_tensorcnt(0);
#else
        asm volatile("s_wait_tensorcnt 0x0" ::: "memory");
#endif
        asm volatile("" ::: "memory");          // keep LDS reads below the TDM fill
    }
#else
    // fallback: cooperative copy with same pad mapping dest = k + k/32
    for (int k = threadIdx.x; k < 16896; k += 256)
        lds[k + (k >> 5)] = ws[WS_SUMS + k];
#endif
    __syncthreads();

    float acc = 0.f;
    const int stride = gridDim.x * 256;
    for (int i = blockIdx.x * 256 + threadIdx.x; i < n; i += stride) {
        __builtin_prefetch((const void*)(x4 + (size_t)(i + stride) * 8), 0, 1);
        float c[DIM];
        load_point(x4, (size_t)i, c);
        float inv = inv_norm(c);
        int seg = sb[i] * NL + lab[i];
        const float* mu = &lds[seg * 33];       // padded stride: bank = (33s+d)%64
        float d1 = 0.f;
#pragma unroll
        for (int d = 0; d < DIM; ++d) d1 += fabsf(mu[d] - c[d] * inv);
        float h = fmaxf(d1 - DELTA_V, 0.f);
        acc += h * h * lds[16896 + seg + (seg >> 5)];   // padded scale slot
    }

    red[threadIdx.x] = acc; __syncthreads();
    for (int off = 128; off > 0; off >>= 1) {
        if (threadIdx.x < off) red[threadIdx.x] += red[threadIdx.x + off];
        __syncthreads();
    }
    if (threadIdx.x == 0) atomicAdd(pull_acc, red[0]);
}

// ------------------------------------------------------------ finalize ------
__global__ void cil_finalize(const float* __restrict__ ws, float* __restrict__ out) {
    float push = 0.f;
    for (int b = 0; b < NB; ++b) {
        float M = ws[WS_M + b];
        if (M > 1.f) push += ws[WS_PUSHB + b] / (M * (M - 1.0f));
    }
    float bc = ws[WS_BCOUNT];
    out[0] = (ws[WS_PULL] + push) / (bc > 0.f ? bc : 1.0f);   // ALPHA = 1
}

// ------------------------------------------------------------- launch -------
extern "C" void kernel_launch(void* const* d_in, const int* in_sizes, int n_in,
                              void* d_out, int out_size, void* d_ws, size_t ws_size,
                              hipStream_t stream) {
    const float4* x4  = (const float4*)d_in[0];
    const int*    lab = (const int*)d_in[1];
    const int*    sb  = (const int*)d_in[2];
    float* ws  = (float*)d_ws;
    float* out = (float*)d_out;
    const int n = in_sizes[1];                  // N points (labels count)

    cil_init     <<<(WS_TOTAL + 255) / 256, 256, 0, stream>>>(ws);
    cil_pass1    <<<512, 256, 0, stream>>>(x4, lab, sb, ws, n);
    cil_centroids<<<1, 512, 0, stream>>>(ws);
    cil_push     <<<128, 256, 0, stream>>>(ws);
    cil_pull     <<<512, 256, 0, stream>>>(x4, lab, sb, ws, ws + WS_PULL, n);
    cil_finalize <<<1, 1, 0, stream>>>(ws, out);
}